// HDTBLUT_87454124081252
// MI455X (gfx1250) — compile-verified
//
#include <hip/hip_runtime.h>
#include <hip/hip_bf16.h>
#include <stdint.h>

// HDTB 4D-LUT 2x super-resolution, fully fused (4 ktypes x 4 rotations).
// img: int32 [4,1,1024,1024], values in [0,16)
// w{h,d,t,b}: float32 [65536, 4]
// out: float32 [4,1,2048,2048]

#define LRDIM 1024
#define BATCH 4
#define TILE  16
#define HALO  3
#define TW    (TILE + 2 * HALO)   // 22
#define LVL   16

__device__ __forceinline__ int reflect_idx(int m) {
    // jnp.pad reflect (edge not repeated): -k -> k ; S-1+k -> S-1-k
    m = (m < 0) ? -m : m;
    m = (m >= LRDIM) ? (2 * LRDIM - 2 - m) : m;
    return m;
}

__global__ __launch_bounds__(256) void HDTBLUT_fused_kernel(
    const int* __restrict__ img,
    const float4* __restrict__ wh,
    const float4* __restrict__ wd,
    const float4* __restrict__ wt,
    const float4* __restrict__ wb,
    float* __restrict__ out)
{
    __shared__ int tile[TW * TW];

    const int tx    = threadIdx.x;           // 0..15
    const int ty    = threadIdx.y;           // 0..15
    const int tid   = ty * 16 + tx;
    const int q0    = blockIdx.x * TILE;
    const int p0    = blockIdx.y * TILE;
    const int batch = blockIdx.z;
    const size_t ibase = (size_t)batch * LRDIM * LRDIM;

    // ---- Stage (TILE+6)^2 halo tile into LDS via CDNA5 async global->LDS ----
    // Per-lane LDS destination + per-lane (reflected) global source address.
    #pragma unroll
    for (int k = 0; k < 2; ++k) {
        int e = tid + k * 256;
        if (e < TW * TW) {
            int lp = e / TW;
            int lq = e - lp * TW;
            int gy = reflect_idx(p0 - HALO + lp);
            int gx = reflect_idx(q0 - HALO + lq);
            const int* gptr = img + ibase + (size_t)gy * LRDIM + gx;
            unsigned lds_off = (unsigned)(size_t)(&tile[e]);   // low 32b = LDS offset
            uint64_t gaddr   = (uint64_t)(uintptr_t)gptr;
            asm volatile("global_load_async_to_lds_b32 %0, %1, off"
                         :: "v"(lds_off), "v"(gaddr)
                         : "memory");
        }
    }
    asm volatile("s_wait_asynccnt 0" ::: "memory");
    __syncthreads();

    // ---- 16 fused passes ----
    // Tap offsets per ktype (rotated-frame (dy,dx)); order h, d, t, b.
    constexpr int OFF[4][4][2] = {
        {{0,0},{0,1},{0,2},{0,3}},   // h
        {{0,0},{1,1},{2,2},{3,3}},   // d
        {{0,0},{2,1},{3,1},{3,2}},   // t
        {{0,0},{1,2},{1,3},{2,3}},   // b
    };
    const float4* __restrict__ wptr[4] = { wh, wd, wt, wb };

    // corner = tile origin of this thread's 7x7 neighborhood -> non-negative DS offsets
    const int* corner = &tile[ty * TW + tx];

    float a00 = 0.f, a01 = 0.f, a10 = 0.f, a11 = 0.f;

    #pragma unroll
    for (int r = 0; r < 4; ++r) {
        #pragma unroll
        for (int k = 0; k < 4; ++k) {
            int t[4];
            #pragma unroll
            for (int s = 0; s < 4; ++s) {
                const int dy = OFF[k][s][0], dx = OFF[k][s][1];
                int u, v;
                if      (r == 0) { u =  dy; v =  dx; }
                else if (r == 1) { u =  dx; v = -dy; }
                else if (r == 2) { u = -dy; v = -dx; }
                else             { u = -dx; v =  dy; }
                t[s] = corner[(u + HALO) * TW + (v + HALO)];
            }
            const int idx = ((t[0] * LVL + t[1]) * LVL + t[2]) * LVL + t[3];
            const float4 w = wptr[k][idx];
            // sub-pixel permutation from inverse rotation of the 2x HR grid
            if      (r == 0) { a00 += w.x; a01 += w.y; a10 += w.z; a11 += w.w; }
            else if (r == 1) { a01 += w.x; a11 += w.y; a00 += w.z; a10 += w.w; }
            else if (r == 2) { a11 += w.x; a10 += w.y; a01 += w.z; a00 += w.w; }
            else             { a10 += w.x; a00 += w.y; a11 += w.z; a01 += w.w; }
        }
    }

    // ---- Write 2x2 HR block once ----
    const int p = p0 + ty, q = q0 + tx;
    float* obase = out + (size_t)batch * (2 * LRDIM) * (2 * LRDIM);
    float2 r0 = make_float2(a00 * 0.25f, a01 * 0.25f);
    float2 r1 = make_float2(a10 * 0.25f, a11 * 0.25f);
    *(float2*)(obase + (size_t)(2 * p)     * (2 * LRDIM) + 2 * q) = r0;
    *(float2*)(obase + (size_t)(2 * p + 1) * (2 * LRDIM) + 2 * q) = r1;
}

extern "C" void kernel_launch(void* const* d_in, const int* in_sizes, int n_in,
                              void* d_out, int out_size, void* d_ws, size_t ws_size,
                              hipStream_t stream) {
    const int*    img = (const int*)d_in[0];
    const float4* wh  = (const float4*)d_in[1];
    const float4* wd  = (const float4*)d_in[2];
    const float4* wt  = (const float4*)d_in[3];
    const float4* wb  = (const float4*)d_in[4];
    float* out = (float*)d_out;

    dim3 grid(LRDIM / TILE, LRDIM / TILE, BATCH);   // 64 x 64 x 4
    dim3 block(16, 16, 1);                          // 256 threads = 8 wave32
    hipLaunchKernelGGL(HDTBLUT_fused_kernel, grid, block, 0, stream,
                       img, wh, wd, wt, wb, out);
}